// Rocket_14723147890713
// MI455X (gfx1250) — compile-verified
//
#include <hip/hip_runtime.h>

// ROCKET-style dilated conv1d + steep-sigmoid PPV, fp32-exact via
// V_WMMA_F32_16X16X4_F32 (CDNA5 fp32 matrix pipe, wave32).
//
// GEMM view per (batch b, dilation d):
//   Y[k, t] = sum_{j=0..8, c=0..3} W[k,c,j] * x[b,c, t + (j-4)*d]
// K-dim = 36 = 9 WMMA steps of K=4, one step per tap j (4 channels/step).

typedef float v2f __attribute__((ext_vector_type(2)));
typedef float v8f __attribute__((ext_vector_type(8)));

#define SEQ_LEN   4096
#define KS        9
#define NKER      256
#define NCH       4
#define NDIL      10
#define NFEAT     (NKER * NDIL)

// sigmoid(50*y) = 1 / (1 + exp2(y * (-50 * log2(e))));  v_exp_f32 is base-2.
#define NEG_SIGC_LOG2E  (-72.13475204444817f)   // -50 * log2(e)

__device__ __constant__ int c_dils[NDIL] = {1, 2, 4, 8, 16, 32, 64, 128, 256, 511};

__global__ void __launch_bounds__(256)
rocket_ppv_f32wmma(const float* __restrict__ x,   // [32, 4, 4096]
                   const float* __restrict__ W,   // [256, 4, 9]
                   float* __restrict__ out)       // [32, 2560]
{
    // x[b] staged time-major: xs[t*4 + c], 64 KB (WGP has 320 KB LDS)
    __shared__ float xs[SEQ_LEN * NCH];
    __shared__ float wsum[8 * 16];     // per-wave partial kernel sums

    const int ktile = blockIdx.x;      // 0..15 : 16 kernels per tile
    const int di    = blockIdx.y;      // 0..9
    const int b     = blockIdx.z;      // 0..31
    const int d     = c_dils[di];

    const int tid  = threadIdx.x;
    const int lane = tid & 31;
    const int wave = tid >> 5;         // 0..7
    const int hf   = lane >> 4;        // half-wave (K/M split per ISA layout)
    const int n    = lane & 15;        // N column within tile

    // ---- stage x[b] into LDS, transpose [c][t] -> [t][c] ----
    for (int t = tid; t < SEQ_LEN; t += 256) {
        float4 v;
        v.x = x[(b * NCH + 0) * SEQ_LEN + t];
        v.y = x[(b * NCH + 1) * SEQ_LEN + t];
        v.z = x[(b * NCH + 2) * SEQ_LEN + t];
        v.w = x[(b * NCH + 3) * SEQ_LEN + t];
        *(float4*)&xs[t * NCH] = v;    // 16B aligned
    }
    __syncthreads();

    // ---- A fragments: 16x4 fp32 per tap j ----
    // ISA layout 16x4 A: VGPR0 = K0 (lanes 0-15) | K2 (lanes 16-31),
    //                    VGPR1 = K1 | K3;  M = lane & 15.
    // K-in-step = channel, so a[0] = W[m, 2*hf, j], a[1] = W[m, 2*hf+1, j].
    v2f A[KS];
    {
        const int m  = ktile * 16 + n;
        const int c0 = 2 * hf;
        #pragma unroll
        for (int j = 0; j < KS; ++j) {
            A[j].x = W[(m * NCH + c0    ) * KS + j];
            A[j].y = W[(m * NCH + c0 + 1) * KS + j];
        }
    }

    v8f acc = {0.f, 0.f, 0.f, 0.f, 0.f, 0.f, 0.f, 0.f};  // running sigmoid sums
    const int halo = (KS / 2) * d;     // 4*d : max tap reach on either side

    // Each wave owns time tiles wave, wave+8, ... (256 tiles of 16 total)
    for (int tt = wave; tt < SEQ_LEN / 16; tt += 8) {
        const int t0 = tt * 16;
        v8f c8 = {0.f, 0.f, 0.f, 0.f, 0.f, 0.f, 0.f, 0.f};

        if (t0 >= halo && (t0 + 15 + halo) < SEQ_LEN) {
            // ---- fast path: whole 9-tap window in-bounds ----
            #pragma unroll
            for (int j = 0; j < KS; ++j) {
                // B fragment 4x16: VGPR0 = K0|K2 rows, VGPR1 = K1|K3 rows
                const int t = t0 + n + (j - KS / 2) * d;
                v2f bv = *(const v2f*)&xs[t * NCH + 2 * hf];  // ds_load_b64
                c8 = __builtin_amdgcn_wmma_f32_16x16x4_f32(
                         false, A[j], false, bv, (short)0, c8, false, false);
            }
        } else {
            // ---- edge path: branch-free zero-padding ----
            // Clamp the address (always-legal load) and zero via cndmask-mask:
            // no EXEC writes, all 9 ds_loads can batch ahead of the WMMA chain.
            #pragma unroll
            for (int j = 0; j < KS; ++j) {
                const int t = t0 + n + (j - KS / 2) * d;
                const bool inb = (unsigned)t < (unsigned)SEQ_LEN;
                const int tc = inb ? t : 0;
                const float msk = inb ? 1.0f : 0.0f;
                v2f bv = *(const v2f*)&xs[tc * NCH + 2 * hf];
                bv.x *= msk;
                bv.y *= msk;
                c8 = __builtin_amdgcn_wmma_f32_16x16x4_f32(
                         false, A[j], false, bv, (short)0, c8, false, false);
            }
        }

        // fused steep sigmoid + accumulate (C layout: VGPR v -> M = v + 8*hf).
        // One pk-mul pass, then hardware exp2 + rcp on the trans pipe.
        #pragma unroll
        for (int v = 0; v < 8; ++v) {
            float e = __builtin_amdgcn_exp2f(c8[v] * NEG_SIGC_LOG2E);
            acc[v] += __builtin_amdgcn_rcpf(1.0f + e);
        }
    }

    // ---- deterministic reduction ----
    // 1) sum over the 16 N-columns within each half-wave (xor shuffles stay
    //    inside each 16-lane group for offsets 1,2,4,8)
    #pragma unroll
    for (int v = 0; v < 8; ++v) {
        float s = acc[v];
        s += __shfl_xor(s, 1, 32);
        s += __shfl_xor(s, 2, 32);
        s += __shfl_xor(s, 4, 32);
        s += __shfl_xor(s, 8, 32);
        acc[v] = s;
    }
    // 2) lane n==0 of each half-wave publishes its 8 kernel sums
    if (n == 0) {
        #pragma unroll
        for (int v = 0; v < 8; ++v) {
            wsum[wave * 16 + hf * 8 + v] = acc[v];
        }
    }
    __syncthreads();
    // 3) fixed-order sum across the 8 waves -> replay-deterministic
    if (tid < 16) {
        float s = 0.f;
        #pragma unroll
        for (int w = 0; w < 8; ++w) s += wsum[w * 16 + tid];
        out[b * NFEAT + di * NKER + ktile * 16 + tid] = s * (1.0f / SEQ_LEN);
    }
}

extern "C" void kernel_launch(void* const* d_in, const int* in_sizes, int n_in,
                              void* d_out, int out_size, void* d_ws, size_t ws_size,
                              hipStream_t stream) {
    (void)in_sizes; (void)n_in; (void)d_ws; (void)ws_size; (void)out_size;
    const float* x = (const float*)d_in[0];   // [32, 4, 4096]
    const float* W = (const float*)d_in[1];   // [256, 4, 9]
    float* out = (float*)d_out;               // [32, 2560]

    dim3 grid(NKER / 16, NDIL, 32);           // (k-tile, dilation, batch)
    dim3 block(256);                           // 8 waves (wave32)
    rocket_ppv_f32wmma<<<grid, block, 0, stream>>>(x, W, out);
}